// CausalFFTConv_1348619731136
// MI455X (gfx1250) — compile-verified
//
#include <hip/hip_runtime.h>
#include <hip/hip_bf16.h>

typedef __attribute__((ext_vector_type(2))) float v2f;
typedef __attribute__((ext_vector_type(8))) float v8f;

#define BB 4
#define TT 8192
#define DD 1024
#define LC 512                 // timesteps per chunk
#define NC (TT / LC)           // 16 chunks
#define NSUB (LC / 16)         // 32 sub-tiles of 16 timesteps per chunk

// float offsets into workspace
#define OFF_ERC 0                          // e^{+a r} cos(w r), [16][D]
#define OFF_ERS (OFF_ERC + 16 * DD)        // e^{+a r} sin(w r)
#define OFF_DRC (OFF_ERS + 16 * DD)        // e^{-a r} cos(w r)
#define OFF_DRS (OFF_DRC + 16 * DD)        // e^{-a r} sin(w r)
#define OFF_ZLC (OFF_DRS + 16 * DD)        // z^LC per d: (re, im), [D][2]
#define OFF_E   (OFF_ZLC + 2 * DD)         // chunk-local end states, [B][NC][D][2]
#define OFF_HIN (OFF_E + BB * NC * DD * 2) // chunk incoming states,  [B][NC][D][2]

// -----------------------------------------------------------------------------
// Pass 0: per-channel twiddle/decay tables. z_d = exp(-|decay_d| + i*freq_d).
// -----------------------------------------------------------------------------
__global__ void build_tables(const float* __restrict__ decay,
                             const float* __restrict__ freq,
                             float* __restrict__ ws) {
  int d = blockIdx.x * blockDim.x + threadIdx.x;
  if (d >= DD) return;
  float a = fabsf(decay[d]);
  float w = freq[d];
  for (int r = 0; r < 16; ++r) {
    float ar = a * (float)r;
    float wr = w * (float)r;
    float c = cosf(wr), s = sinf(wr);
    float ep = expf(ar), em = expf(-ar);
    ws[OFF_ERC + r * DD + d] = ep * c;   // z^{-r} = e^{ar}(cos - i sin)
    ws[OFF_ERS + r * DD + d] = ep * s;
    ws[OFF_DRC + r * DD + d] = em * c;   // z^{+r} = e^{-ar}(cos + i sin)
    ws[OFF_DRS + r * DD + d] = em * s;
  }
  float aL = a * (float)LC, wL = w * (float)LC;
  float emL = expf(-aL);
  ws[OFF_ZLC + 2 * d + 0] = emL * cosf(wL);
  ws[OFF_ZLC + 2 * d + 1] = emL * sinf(wL);
}

// -----------------------------------------------------------------------------
// Passes 1 & 3: chunked scan. One wave owns a (b, chunk, 16-channel) tile-
// sequence and walks LC timesteps in 16x16 (time x channel) WMMA sub-tiles.
//   Q[r,d] = z_d*h + sum_{s<=r} z_d^{-s} x[s,d]   (prefix via ones-triangular L)
//   y[r,d] = Re(z_d^r * Q[r,d]),  h' = z^{15} * Q[15,d]
// The carry term z*h is folded into u[0] (column 0 of L is all-ones), so the
// WMMA accumulators start from a plain zero vector (inline-0 SRC2 eligible).
// MODE 0: zero carry, record chunk end state E; x read with RT hints so the
//         128MB of x parks in the 192MB L2 for the second pass.
// MODE 1: carry = Hin, write y; x loads and y stores are non-temporal so the
//         streamed output does not evict x between the two passes.
// -----------------------------------------------------------------------------
template <int MODE>
__global__ void __launch_bounds__(256) scan_pass(const float* __restrict__ x,
                                                 float* __restrict__ y,
                                                 float* __restrict__ ws) {
  const int lane = threadIdx.x & 31;
  const int widx = threadIdx.x >> 5;
  const int g = blockIdx.x * (blockDim.x >> 5) + widx;  // 0 .. B*NC*(D/16)-1
  const int dtile = g & 63;                             // D/16 = 64
  const int chunk = (g >> 6) & (NC - 1);
  const int b = g >> 10;
  const int n = lane & 15;   // channel column within tile / matrix N (and A's M)
  const int half = lane >> 4;
  const int d = dtile * 16 + n;

  const float* erc = ws + OFF_ERC;
  const float* ers = ws + OFF_ERS;
  const float* drc = ws + OFF_DRC;
  const float* drs = ws + OFF_DRS;

  // B-operand scale factors: element (k = 2*half + p) of slice j is time r = 4j+k.
  v2f bscc[4], bscs[4];
#pragma unroll
  for (int j = 0; j < 4; ++j)
#pragma unroll
    for (int p = 0; p < 2; ++p) {
      int r = 4 * j + 2 * half + p;
      bscc[j][p] = erc[r * DD + d];
      bscs[j][p] = ers[r * DD + d];
    }
  // Output unscale factors: C/D VGPR v holds row m = v + 8*half.
  float oc[8], os[8];
#pragma unroll
  for (int v = 0; v < 8; ++v) {
    int m = v + 8 * half;
    oc[v] = drc[m * DD + d];
    os[v] = drs[m * DD + d];
  }
  const float d1c = drc[1 * DD + d], d1s = drs[1 * DD + d];     // z^1
  const float d15c = drc[15 * DD + d], d15s = drs[15 * DD + d]; // z^15

  // Constant lower-triangular ones matrix L, split into 16x4 A-slices.
  // A layout: lane holds m = lane&15, k = 2*half + p.
  v2f A[4];
#pragma unroll
  for (int j = 0; j < 4; ++j)
#pragma unroll
    for (int p = 0; p < 2; ++p) {
      int k = 2 * half + p;
      A[j][p] = ((4 * j + k) <= n) ? 1.0f : 0.0f;
    }
  // 1.0 on the lanes holding B element k=0 (half==0, p==0): carry injection mask.
  const float sel0 = (half == 0) ? 1.0f : 0.0f;

  // Carry (complex state at t0-1), identical across the two half-waves.
  float hr = 0.0f, hi = 0.0f;
  if (MODE == 1) {
    int o = ((b * NC + chunk) * DD + d) * 2;
    hr = ws[OFF_HIN + o + 0];
    hi = ws[OFF_HIN + o + 1];
  }

  const float* xb = x + ((size_t)b * TT + (size_t)chunk * LC) * DD + d;
  float* yb = y + ((size_t)b * TT + (size_t)chunk * LC) * DD + d;

  for (int it = 0; it < NSUB; ++it) {
    const float* xt = xb + (size_t)it * 16 * DD;

    // Load x tile and scale into complex u = x * z^{-r}  (B operands).
    v2f Bre[4], Bim[4];
#pragma unroll
    for (int j = 0; j < 4; ++j)
#pragma unroll
      for (int p = 0; p < 2; ++p) {
        int r = 4 * j + 2 * half + p;
        float xv = (MODE == 1) ? __builtin_nontemporal_load(xt + (size_t)r * DD)
                               : xt[(size_t)r * DD];
        Bre[j][p] = xv * bscc[j][p];
        Bim[j][p] = -xv * bscs[j][p];
      }

    // Fold carry into u[0]: Q[r] = z*h + prefix  ==  prefix with u[0] += z*h.
    float zhr = d1c * hr - d1s * hi;
    float zhi = d1c * hi + d1s * hr;
    Bre[0][0] = fmaf(sel0, zhr, Bre[0][0]);
    Bim[0][0] = fmaf(sel0, zhi, Bim[0][0]);

    // Q = L @ u  via 4 accumulating K=4 WMMAs per complex component.
    v8f accr = {0.f, 0.f, 0.f, 0.f, 0.f, 0.f, 0.f, 0.f};
    v8f acci = {0.f, 0.f, 0.f, 0.f, 0.f, 0.f, 0.f, 0.f};
#pragma unroll
    for (int j = 0; j < 4; ++j) {
      accr = __builtin_amdgcn_wmma_f32_16x16x4_f32(false, A[j], false, Bre[j],
                                                   (short)0, accr, false, false);
      acci = __builtin_amdgcn_wmma_f32_16x16x4_f32(false, A[j], false, Bim[j],
                                                   (short)0, acci, false, false);
    }

    if (MODE == 1) {
      // y[r,d] = Re(z^r * Q) = e^{-ar}(cos(wr)*Qre - sin(wr)*Qim)
#pragma unroll
      for (int v = 0; v < 8; ++v) {
        int m = v + 8 * half;
        float yv = oc[v] * accr[v] - os[v] * acci[v];
        __builtin_nontemporal_store(yv, yb + (size_t)(it * 16 + m) * DD);
      }
    }

    // Carry update: h' = z^{15} * Q[15].  Row 15 lives in VGPR 7, lanes 16-31.
    float q15r = __shfl(accr[7], 16 + n, 32);
    float q15i = __shfl(acci[7], 16 + n, 32);
    hr = d15c * q15r - d15s * q15i;
    hi = d15c * q15i + d15s * q15r;
  }

  if (MODE == 0 && half == 0) {
    int o = ((b * NC + chunk) * DD + d) * 2;
    ws[OFF_E + o + 0] = hr;
    ws[OFF_E + o + 1] = hi;
  }
}

// -----------------------------------------------------------------------------
// Pass 2: tiny per-(b,d) scan over NC=16 chunk states: Hin[c] = S_{c-1},
// S_c = z^{LC} * S_{c-1} + E_c.
// -----------------------------------------------------------------------------
__global__ void combine_pass(float* __restrict__ ws) {
  int idx = blockIdx.x * blockDim.x + threadIdx.x;
  if (idx >= BB * DD) return;
  int b = idx / DD, d = idx % DD;
  float zr = ws[OFF_ZLC + 2 * d + 0];
  float zi = ws[OFF_ZLC + 2 * d + 1];
  float hr = 0.0f, hi = 0.0f;
  for (int c = 0; c < NC; ++c) {
    int o = ((b * NC + c) * DD + d) * 2;
    ws[OFF_HIN + o + 0] = hr;
    ws[OFF_HIN + o + 1] = hi;
    float er = ws[OFF_E + o + 0];
    float ei = ws[OFF_E + o + 1];
    float nhr = zr * hr - zi * hi + er;
    float nhi = zr * hi + zi * hr + ei;
    hr = nhr;
    hi = nhi;
  }
}

extern "C" void kernel_launch(void* const* d_in, const int* in_sizes, int n_in,
                              void* d_out, int out_size, void* d_ws, size_t ws_size,
                              hipStream_t stream) {
  const float* x = (const float*)d_in[0];
  const float* decay = (const float*)d_in[1];
  const float* freq = (const float*)d_in[2];
  float* y = (float*)d_out;
  float* ws = (float*)d_ws;

  build_tables<<<(DD + 255) / 256, 256, 0, stream>>>(decay, freq, ws);

  const int waves = BB * NC * (DD / 16);  // 4096 tile-sequences
  const int blocks = waves / 8;           // 8 waves (256 threads) per block
  scan_pass<0><<<blocks, 256, 0, stream>>>(x, y, ws);
  combine_pass<<<(BB * DD + 255) / 256, 256, 0, stream>>>(ws);
  scan_pass<1><<<blocks, 256, 0, stream>>>(x, y, ws);
}